// SSIMLoss_11278584119529
// MI455X (gfx1250) — compile-verified
//
#include <hip/hip_runtime.h>

typedef float v2f __attribute__((ext_vector_type(2)));
typedef float v8f __attribute__((ext_vector_type(8)));

#define IMG_H 320
#define IMG_W 320
#define NIMG  128
#define TS    16
#define TPR   (IMG_W / TS)           // 20 tiles per row
#define TPI   (TPR * (IMG_H / TS))   // 400 tiles per image
#define NTILES (NIMG * TPI)          // 51200
#define WAVES_PER_BLOCK 8
#define HALO_ROWS 26
#define HALO_STR  28                 // padded halo stride (cols 26,27 zeroed)
#define MID_ROWS  28                 // mid rows 26,27 zeroed
#define PER_WAVE_F (HALO_ROWS*HALO_STR*2 + MID_ROWS*TS)  // 1456+448 = 1904 floats

// Normalized 11-tap Gaussian, sigma=1.5 (matches reference exactly in fp32)
__device__ const float WGc[11] = {
  0.00102838f, 0.00759876f, 0.03600077f, 0.10936069f, 0.21300553f,
  0.26601172f, 0.21300553f, 0.10936069f, 0.03600077f, 0.00759876f,
  0.00102838f
};

extern "C" __global__ void ssim_init(double* acc) {
  if (threadIdx.x == 0 && blockIdx.x == 0) acc[0] = 0.0;
}

extern "C" __global__ __launch_bounds__(256)
void ssim_main(const float* __restrict__ pred, const float* __restrict__ targ,
               double* __restrict__ accum)
{
  __shared__ float smem[WAVES_PER_BLOCK * PER_WAVE_F];
  __shared__ float wgS[32];

  const int tid  = threadIdx.x;
  const int lane = tid & 31;
  const int wid  = tid >> 5;

  // Stage weights into LDS so banded operands can index dynamically.
  if (tid < 32) wgS[tid] = (tid < 11) ? WGc[tid] : 0.0f;
  __syncthreads();

  const int tile = blockIdx.x * WAVES_PER_BLOCK + wid;  // NTILES divisible by 8
  const int img  = tile / TPI;
  const int tin  = tile % TPI;
  const int r0   = (tin / TPR) * TS;
  const int c0   = (tin % TPR) * TS;

  float* pL = smem + wid * PER_WAVE_F;        // 26x28 pred halo
  float* tL = pL + HALO_ROWS * HALO_STR;      // 26x28 target halo
  float* mL = tL + HALO_ROWS * HALO_STR;      // 28x16 mid tile

  const float* pBase = pred + (size_t)img * (IMG_H * IMG_W);
  const float* tBase = targ + (size_t)img * (IMG_H * IMG_W);

  // ---- load halo tiles with zero padding (reference uses zero-pad conv) ----
  for (int idx = lane; idx < HALO_ROWS * HALO_STR; idx += 32) {
    const int rr = idx / HALO_STR, cc = idx % HALO_STR;
    const int gr = r0 - 5 + rr,    gc = c0 - 5 + cc;
    float pv = 0.0f, tv = 0.0f;
    if (cc < 26 && gr >= 0 && gr < IMG_H && gc >= 0 && gc < IMG_W) {
      pv = pBase[gr * IMG_W + gc];
      tv = tBase[gr * IMG_W + gc];
    }
    pL[idx] = pv;
    tL[idx] = tv;
  }
  // zero mid pad rows 26,27 (32 entries, one per lane)
  mL[26 * TS + lane] = 0.0f;
  asm volatile("s_wait_dscnt 0" ::: "memory");
  __builtin_amdgcn_wave_barrier();

  const int m16 = lane & 15;   // N-index for B/C/D, M-index for A
  const int hi  = lane >> 4;   // lane half selects K pair {0,1} vs {2,3}

  // ---- banded weight operand, band[k][n] = w[k-n] (same for H-B and V-A) ----
  v2f band[7];
#pragma unroll
  for (int kc = 0; kc < 7; ++kc) {
    const int k = 4 * kc + 2 * hi;
    band[kc].x = wgS[(k     - m16) & 31];  // indices 11..31 hold 0
    band[kc].y = wgS[(k + 1 - m16) & 31];
  }

  v8f res[5];

#pragma unroll
  for (int ch = 0; ch < 5; ++ch) {
    // ---- horizontal pass: two 16-row strips cover mid rows 0..25 ----
#pragma unroll
    for (int rt = 0; rt < 2; ++rt) {
      const int roff = rt * 10;          // rows 0..15 then 10..25
      v8f mid = {};
#pragma unroll
      for (int kc = 0; kc < 7; ++kc) {
        const int col = 4 * kc + 2 * hi;
        const int row = roff + m16;
        const float p0 = pL[row * HALO_STR + col];
        const float p1 = pL[row * HALO_STR + col + 1];
        const float t0 = tL[row * HALO_STR + col];
        const float t1 = tL[row * HALO_STR + col + 1];
        v2f a;
        switch (ch) {
          case 0:  a.x = p0;      a.y = p1;      break;  // mu_x path
          case 1:  a.x = t0;      a.y = t1;      break;  // mu_y path
          case 2:  a.x = p0 * p0; a.y = p1 * p1; break;  // E[x^2]
          case 3:  a.x = t0 * t0; a.y = t1 * t1; break;  // E[y^2]
          default: a.x = p0 * t0; a.y = p1 * t1; break;  // E[xy]
        }
        mid = __builtin_amdgcn_wmma_f32_16x16x4_f32(
            false, a, false, band[kc], (short)0, mid, false, false);
      }
      // D layout: VGPR v holds row (v + 8*hi), col m16
#pragma unroll
      for (int v = 0; v < 8; ++v) {
        const int row = roff + v + 8 * hi;
        mL[row * TS + m16] = mid[v];
      }
      asm volatile("s_wait_dscnt 0" ::: "memory");
      __builtin_amdgcn_wave_barrier();
    }

    // ---- vertical pass: Out = band(16x28) x Mid(28x16) ----
    v8f acc = {};
#pragma unroll
    for (int kc = 0; kc < 7; ++kc) {
      const int k = 4 * kc + 2 * hi;
      v2f b;
      b.x = mL[k * TS + m16];
      b.y = mL[(k + 1) * TS + m16];
      acc = __builtin_amdgcn_wmma_f32_16x16x4_f32(
          false, band[kc], false, b, (short)0, acc, false, false);
    }
    res[ch] = acc;
    asm volatile("s_wait_dscnt 0" ::: "memory");
    __builtin_amdgcn_wave_barrier();
  }

  // ---- pointwise SSIM map + per-lane partial sum (8 pixels per lane) ----
  const float c1 = 1e-4f, c2 = 9e-4f;
  float s = 0.0f;
#pragma unroll
  for (int v = 0; v < 8; ++v) {
    const float mx  = res[0][v], my  = res[1][v];
    const float ex2 = res[2][v], ey2 = res[3][v], exy = res[4][v];
    const float mx2 = mx * mx, my2 = my * my, mxy = mx * my;
    const float sx  = ex2 - mx2;
    const float sy  = ey2 - my2;
    const float sxy = exy - mxy;
    const float num = (2.0f * mxy + c1) * (2.0f * sxy + c2);
    const float den = (mx2 + my2 + c1) * (sx + sy + c2);
    s += num / den;
  }
  // wave32 reduction
  for (int off = 16; off > 0; off >>= 1)
    s += __shfl_xor(s, off, 32);
  if (lane == 0)
    atomicAdd(accum, (double)s);
}

extern "C" __global__ void ssim_fin(const double* acc, float* out) {
  if (threadIdx.x == 0 && blockIdx.x == 0) {
    const double n = (double)NIMG * IMG_H * IMG_W;  // 13,107,200
    out[0] = (float)(1.0 - acc[0] / n);
  }
}

extern "C" void kernel_launch(void* const* d_in, const int* in_sizes, int n_in,
                              void* d_out, int out_size, void* d_ws, size_t ws_size,
                              hipStream_t stream) {
  const float* pred = (const float*)d_in[0];
  const float* targ = (const float*)d_in[1];
  double* ws = (double*)d_ws;       // 8-byte fp64 accumulator
  float* out = (float*)d_out;

  ssim_init<<<1, 1, 0, stream>>>(ws);
  ssim_main<<<NTILES / WAVES_PER_BLOCK, 32 * WAVES_PER_BLOCK, 0, stream>>>(pred, targ, ws);
  ssim_fin<<<1, 1, 0, stream>>>(ws, out);
}